// MultiHeadAttentionQuantum_65481071400826
// MI455X (gfx1250) — compile-verified
//
#include <hip/hip_runtime.h>

// MI455X / gfx1250, wave32. Quantum layer reduced analytically to cosine
// prefix products; attention + output projection with fp32 WMMA
// (V_WMMA_F32_16X16X4_F32). Problem is latency/bandwidth bound (~2.2 GFLOP,
// ~4MB traffic vs 23.3 TB/s HBM), so exact fp32 matmul is free. Flash
// softmax on TRANSPOSED score tiles (in-lane reductions, 2 cross-half
// permutes per tile); P^T feeds the PV WMMA directly from registers via a
// K-permutation trick. LDS staging uses GLOBAL_LOAD_ASYNC_TO_LDS_B128
// (ASYNCcnt path); exps use raw v_exp_f32.

typedef float v2f __attribute__((ext_vector_type(2)));
typedef float v4f __attribute__((ext_vector_type(4)));
typedef float v8f __attribute__((ext_vector_type(8)));

#define BQ   8
#define SQ   1024
#define EMB  64
#define HQ   8
#define DKQ  8
#define NTOK (BQ * SQ * HQ) // 65536 token-heads

__device__ __forceinline__ v8f wmma_f32(v2f a, v2f b, v8f c) {
  // D = A(16x4, f32) * B(4x16, f32) + C(16x16, f32)
  return __builtin_amdgcn_wmma_f32_16x16x4_f32(
      /*neg_a=*/false, a, /*neg_b=*/false, b,
      /*c_mod=*/(short)0, c, /*reuse_a=*/false, /*reuse_b=*/false);
}

// raw v_exp_f32: args are <= 0 here and anything < -126 should flush to 0,
// so ocml's denorm fixup (cmp/cndmask/ldexp per call) is dead weight.
__device__ __forceinline__ float exp2_raw(float x) {
  return __builtin_amdgcn_exp2f(x);
}

// ---------------------------------------------------------------------------
// Kernel 1: quantum layer -> proj[(b*8+h)][s][d]   (cosine prefix products)
//   z0 = c1*...*c7 ; zw = c0*...*cw (w>=1), c_j = cos(x_j + phi_j)
// ---------------------------------------------------------------------------
__global__ void __launch_bounds__(256) qproj_kernel(
    const float* __restrict__ x, const float* __restrict__ phi,
    float* __restrict__ proj) {
  int th = blockIdx.x * 256 + threadIdx.x; // exactly 65536 threads
  const float* xs = x + (size_t)th * DKQ;  // E = H*DK, head-contiguous
  float c[8];
#pragma unroll
  for (int j = 0; j < 8; ++j) c[j] = __cosf(xs[j] + phi[j]);
  float z[8];
  float acc = c[0];
#pragma unroll
  for (int w = 1; w < 8; ++w) { acc *= c[w]; z[w] = acc; }
  float suf = c[1];
#pragma unroll
  for (int j = 2; j < 8; ++j) suf *= c[j];
  z[0] = suf;
  int b = th >> 13; // /(S*H)
  int rem = th & 8191;
  int s = rem >> 3;
  int h = rem & 7;
  float* dst = proj + (((size_t)(b * HQ + h) * SQ) + s) * DKQ;
  *(v4f*)dst       = *(v4f*)&z[0];
  *(v4f*)(dst + 4) = *(v4f*)&z[4];
}

// ---------------------------------------------------------------------------
// Kernel 2: flash attention per (b,h).  grid = (64, 8), 256 thr = 8 waves.
// Wave owns query tile qt = blockIdx.y*8 + waveId.  All tiles TRANSPOSED:
//   S^T = Pk * Pq^T   (C layout: M=key=g+8*hi, N=query=lane&15)
//   acc^T = V^T * P^T (C layout: M=d,          N=query=lane&15)
// ---------------------------------------------------------------------------
__global__ void __launch_bounds__(256) attn_kernel(
    const float* __restrict__ proj, float* __restrict__ attn_out) {
  __shared__ float Pl[SQ * DKQ]; // 32 KB: proj for this (b,h)

  const int bh   = blockIdx.x;
  const int tid  = threadIdx.x;
  const int wid  = tid >> 5;
  const int lane = tid & 31;
  const int n    = lane & 15; // query column owned by this lane
  const int hi   = lane >> 4; // which 16-lane half
  const int kb   = hi * 2;    // K sub-offset for f32 A/B operands
  const int d8   = n & 7;

  // async stage of proj[bh] into LDS: memory -> LDS directly (ASYNCcnt),
  // no VGPR round trip.  8 x b128 per thread, 16B aligned.
  {
    const float* Pg = proj + (size_t)bh * (SQ * DKQ);
    // low 32 bits of the generic pointer == LDS byte offset (aperture map)
    const unsigned lds0 = (unsigned)(size_t)(void*)&Pl[0];
#pragma unroll
    for (int i = 0; i < 8; ++i) {
      int idx = (tid + i * 256) * 4;
      unsigned loff = lds0 + (unsigned)idx * 4u;
      const float* gptr = Pg + idx;
      asm volatile("global_load_async_to_lds_b128 %0, %1, off"
                   :: "v"(loff), "v"(gptr)
                   : "memory");
    }
    asm volatile("s_wait_asynccnt 0x0" ::: "memory");
  }
  __syncthreads();

  const int qt = blockIdx.y * 8 + wid; // 0..63
  // fold 1/sqrt(DK) * log2(e) into the query operand: softmax in exp2 units
  const float qscale = 0.35355339059327373f * 1.4426950408889634f;

  // B operand for scores: Pq^T (K=d chunks), loaded once, pre-scaled
  const int rowa = qt * 16 + n;
  v2f bq0 = *(const v2f*)&Pl[rowa * DKQ + kb];
  v2f bq1 = *(const v2f*)&Pl[rowa * DKQ + 4 + kb];
  bq0 *= qscale;
  bq1 *= qscale;

  v8f acc = {};
  float m = -3.0e38f, lsum = 0.0f;

  for (int kt = 0; kt < SQ / 16; ++kt) {
    // A operand for scores: Pk rows
    const int rowb = kt * 16 + n;
    v2f ak0 = *(const v2f*)&Pl[rowb * DKQ + kb];
    v2f ak1 = *(const v2f*)&Pl[rowb * DKQ + 4 + kb];
    v8f s = {};
    s = wmma_f32(ak0, bq0, s);
    s = wmma_f32(ak1, bq1, s);
    // s[g] = score(key = g + 8*hi, query = n), in log2 units

    // V^T A operand for the PV matmul, with chunk->key permutation
    // chunk c covers keys {2c, 2c+1} (low K slots) and {2c+8, 2c+9} (high)
    v2f va[4];
#pragma unroll
    for (int c = 0; c < 4; ++c) {
      int krow = kt * 16 + 2 * c + 8 * hi;
      float x0 = Pl[krow * DKQ + d8];       // pairs 32B apart -> ds_load_2addr
      float x1 = Pl[(krow + 1) * DKQ + d8];
      va[c].x = (n < 8) ? x0 : 0.0f;        // d rows 8..15 are zero padding
      va[c].y = (n < 8) ? x1 : 0.0f;
    }

    // per-query online softmax over this tile's 16 keys (in-lane + 1 permute)
    float tmax = s[0];
#pragma unroll
    for (int g = 1; g < 8; ++g) tmax = fmaxf(tmax, s[g]);
    tmax = fmaxf(tmax, __shfl_xor(tmax, 16, 32));
    float mn   = fmaxf(m, tmax);
    float corr = exp2_raw(m - mn);
    float p[8];
    float rs = 0.0f;
#pragma unroll
    for (int g = 0; g < 8; ++g) {
      p[g] = exp2_raw(s[g] - mn);
      rs += p[g];
    }
    rs += __shfl_xor(rs, 16, 32);
    lsum = lsum * corr + rs;
    m = mn;
#pragma unroll
    for (int g = 0; g < 8; ++g) acc[g] *= corr;

    // acc^T += V^T * P^T : the p registers ARE the B operand under the same
    // key permutation (C-layout VGPR g holds keys g / g+8 across halves).
#pragma unroll
    for (int c = 0; c < 4; ++c) {
      v2f pb;
      pb.x = p[2 * c];
      pb.y = p[2 * c + 1];
      acc = wmma_f32(va[c], pb, acc);
    }
  }

  // acc^T low half: VGPR g, lane n (<16) holds out[query n][d=g]
  float inv = 1.0f / lsum;
  if (lane < 16) {
    v4f o0, o1;
#pragma unroll
    for (int g = 0; g < 4; ++g) o0[g] = acc[g] * inv;
#pragma unroll
    for (int g = 0; g < 4; ++g) o1[g] = acc[4 + g] * inv;
    const int b = bh >> 3, h = bh & 7;
    float* dst = attn_out + ((size_t)(b * SQ + qt * 16 + n)) * EMB + h * DKQ;
    *(v4f*)dst       = o0; // d contiguous -> 2x global_store_b128
    *(v4f*)(dst + 4) = o1;
  }
}

// ---------------------------------------------------------------------------
// Kernel 3: out = attn @ W^T + bias.  [8192,64] x [64,64].
// One 16x16 output tile per wave, K=64 -> 16 WMMA chunks.
// ---------------------------------------------------------------------------
__global__ void __launch_bounds__(256) outproj_kernel(
    const float* __restrict__ a, const float* __restrict__ W,
    const float* __restrict__ bias, float* __restrict__ out) {
  const int tid  = threadIdx.x;
  const int wid  = tid >> 5;
  const int lane = tid & 31;
  const int n    = lane & 15;
  const int hi   = lane >> 4;
  const int kb   = hi * 2;
  const int tileid = blockIdx.x * 8 + wid; // 2048 tiles
  const int tt = tileid >> 2;              // 512 row tiles
  const int te = tileid & 3;               // 4 col tiles

  v8f acc = {};
#pragma unroll
  for (int c = 0; c < 16; ++c) {
    int k0 = c * 4 + kb;
    // A: attn rows.  B[k][e] = W[e][k] -> contiguous pair along k in W rows.
    v2f av = *(const v2f*)&a[(size_t)(tt * 16 + n) * EMB + k0];
    v2f bv = *(const v2f*)&W[(size_t)(te * 16 + n) * EMB + k0];
    acc = wmma_f32(av, bv, acc);
  }
  float bcol = bias[te * 16 + n];
#pragma unroll
  for (int g = 0; g < 8; ++g) {
    int row = tt * 16 + g + 8 * hi;
    out[(size_t)row * EMB + te * 16 + n] = acc[g] + bcol;
  }
}

// ---------------------------------------------------------------------------
extern "C" void kernel_launch(void* const* d_in, const int* in_sizes, int n_in,
                              void* d_out, int out_size, void* d_ws,
                              size_t ws_size, hipStream_t stream) {
  const float* x    = (const float*)d_in[0]; // [8,1024,64]
  const float* phi  = (const float*)d_in[1]; // [8]
  const float* W    = (const float*)d_in[2]; // [64,64]
  const float* bias = (const float*)d_in[3]; // [64]
  float* out = (float*)d_out;                // [8,1024,64]

  float* proj = (float*)d_ws;                 // 64*1024*8 f32 (2 MB)
  float* attn = proj + (size_t)64 * 1024 * 8; // 8192*64  f32 (2 MB)

  qproj_kernel<<<NTOK / 256, 256, 0, stream>>>(x, phi, proj);
  attn_kernel<<<dim3(64, 8), 256, 0, stream>>>(proj, attn);
  outproj_kernel<<<256, 256, 0, stream>>>(attn, W, bias, out);
}